// BambooBase_37752762532668
// MI455X (gfx1250) — compile-verified
//
#include <hip/hip_runtime.h>
#include <hip/hip_bf16.h>

// ---------------------------------------------------------------------------
// BambooBase pairwise Coulomb + dispersion edge kernel for MI455X (gfx1250).
//
// Memory-bound workload: ~208 MB streamed per call vs 23.3 TB/s HBM -> ~9 us.
// Atom tables (charge/c6/r0, 2.4 MB total) are kept L2-resident by marking the
// big edge stream (row/col/dij) and all outputs NON-TEMPORAL, while atom
// gathers use the default RT policy. Edge stream is software-prefetched with
// global_prefetch_b8 (gfx1250 path). All transcendentals use the HW fast
// units (v_exp_f32 / v_log_f32 / v_rcp_f32 / v_rsq_f32 / v_sqrt_f32).
// ---------------------------------------------------------------------------

#define ELE_FACTOR   332.0637f
#define EWALD_F      1.12837917f
#define EWALD_P      0.3275911f
#define EW_A0        0.254829592f
#define EW_A1       -0.284496736f
#define EW_A2        1.421413741f
#define EW_A3       -1.453152027f
#define EW_A4        1.061405429f
#define COUL_BETA    18.7f
#define COUL_R0      2.2f
#define G_EWALD      0.3f
#define R6_OFFSET    8303.765625f      /* 4.5^6 */
#define INV_DISP6    1.0e-6f           /* 1 / 10^6 */

#define PF_DIST 4096                   /* edges of prefetch lead */

__device__ __forceinline__ float fast_rcp(float x)  { return __builtin_amdgcn_rcpf(x);  }
__device__ __forceinline__ float fast_rsq(float x)  { return __builtin_amdgcn_rsqf(x);  }
__device__ __forceinline__ float fast_sqrt(float x) { return __builtin_amdgcn_sqrtf(x); }
__device__ __forceinline__ float fast_exp(float x)  { return __expf(x); }   // v_exp_f32
__device__ __forceinline__ float fast_log(float x)  { return __logf(x); }   // v_log_f32

__global__ void __launch_bounds__(256)
bamboo_edge_kernel(const int*   __restrict__ row,
                   const int*   __restrict__ col,
                   const float* __restrict__ dij,
                   const float* __restrict__ charge,
                   const float* __restrict__ c6,
                   const float* __restrict__ r0,
                   float* __restrict__ out_ecoul,   // [E]
                   float* __restrict__ out_cfij,    // [E,3]
                   float* __restrict__ out_edisp,   // [E]
                   float* __restrict__ out_dfij,    // [E,3]
                   int E)
{
    const int i = blockIdx.x * blockDim.x + threadIdx.x;
    if (i >= E) return;

    // --- software prefetch of the streaming edge data (global_prefetch_b8) ---
    // Speculative (TH=0): dropped silently past the end of the allocation.
    __builtin_prefetch(row + i + PF_DIST, 0, 0);
    __builtin_prefetch(col + i + PF_DIST, 0, 0);
    __builtin_prefetch(dij + 3 * (i + PF_DIST), 0, 0);

    // --- streamed, non-temporal edge loads (keep L2 for the atom tables) ---
    const int ia = __builtin_nontemporal_load(row + i);
    const int ib = __builtin_nontemporal_load(col + i);
    const float dx = __builtin_nontemporal_load(dij + 3 * i + 0);
    const float dy = __builtin_nontemporal_load(dij + 3 * i + 1);
    const float dz = __builtin_nontemporal_load(dij + 3 * i + 2);

    // --- atom gathers: tiny tables (2.4 MB), default RT policy -> L2 hits ---
    const float qi  = charge[ia];
    const float qj  = charge[ib];
    const float c6i = c6[ia];
    const float c6j = c6[ib];
    const float r0i = r0[ia];
    const float r0j = r0[ib];

    // --- shared distance terms ---
    const float r2   = __builtin_fmaf(dx, dx, __builtin_fmaf(dy, dy, dz * dz));
    const float rinv = fast_rsq(r2);        // 1/r
    const float rij  = r2 * rinv;           // r
    const float rinv2 = rinv * rinv;

    // =======================  Coulomb  =======================
    const float prefactor = ELE_FACTOR * qi * qj * rinv;

    // x = (beta/r0)*(r - r0); stable sigmoid & softplus via e = exp(-|x|)
    const float x   = (COUL_BETA / COUL_R0) * (rij - COUL_R0);
    const float ax  = __builtin_fabsf(x);
    const float e   = fast_exp(-ax);
    const float i1e = fast_rcp(1.0f + e);
    const float sig = (x >= 0.0f) ? i1e : e * i1e;                 // sigmoid(x)
    const float sp  = (fmaxf(x, 0.0f) + fast_log(1.0f + e)) * (1.0f / COUL_BETA);
    const float s   = rij * (1.0f / COUL_R0) * fast_rcp(1.0f + sp);

    float ecoul = prefactor * s;
    float fcoul = prefactor * sig * s * s;

    // Ewald short-range correction (erfc Horner)
    const float grij  = G_EWALD * rij;
    const float expm2 = fast_exp(-grij * grij);
    const float t     = fast_rcp(__builtin_fmaf(EWALD_P, grij, 1.0f));
    float poly = __builtin_fmaf(t, EW_A4, EW_A3);
    poly = __builtin_fmaf(t, poly, EW_A2);
    poly = __builtin_fmaf(t, poly, EW_A1);
    poly = __builtin_fmaf(t, poly, EW_A0);
    const float erfc = t * poly * expm2;

    ecoul = __builtin_fmaf(prefactor, erfc - 1.0f, ecoul);
    fcoul = __builtin_fmaf(prefactor,
                           __builtin_fmaf(EWALD_F * grij, expm2, erfc - 1.0f),
                           fcoul);

    const float cscale = fcoul * rinv2;     // fcoul / r^2

    // =======================  Dispersion (D3-CSO)  =======================
    const float c6ij = fast_sqrt(c6i * c6j);
    const float r0ij = 0.5f * (r0i + r0j);

    const float r6    = __builtin_fmaf(r2 * r2, r2, R6_OFFSET);   // r^6 + 4.5^6
    const float invr6 = fast_rcp(r6);
    const float ed    = fast_exp(__builtin_fmaf(-2.5f, r0ij, rij));
    const float i1ed  = fast_rcp(1.0f + ed);
    const float cso   = __builtin_fmaf(0.82f, i1ed, 0.85f);

    const float cr6   = c6ij * invr6;
    float edisp = __builtin_fmaf(-cr6, cso, c6ij * INV_DISP6);

    const float r5    = r2 * r2 * rij;                            // r^5
    const float fdisp = -6.0f * cr6 * r5 * invr6 * cso
                        - cr6 * (0.82f * ed * i1ed * i1ed);
    const float dscale = fdisp * rinv;      // fdisp / r

    // =======================  non-temporal stores  =======================
    __builtin_nontemporal_store(ecoul,       out_ecoul + i);
    __builtin_nontemporal_store(dx * cscale, out_cfij + 3 * i + 0);
    __builtin_nontemporal_store(dy * cscale, out_cfij + 3 * i + 1);
    __builtin_nontemporal_store(dz * cscale, out_cfij + 3 * i + 2);
    __builtin_nontemporal_store(edisp,       out_edisp + i);
    __builtin_nontemporal_store(dx * dscale, out_dfij + 3 * i + 0);
    __builtin_nontemporal_store(dy * dscale, out_dfij + 3 * i + 1);
    __builtin_nontemporal_store(dz * dscale, out_dfij + 3 * i + 2);
}

extern "C" void kernel_launch(void* const* d_in, const int* in_sizes, int n_in,
                              void* d_out, int out_size, void* d_ws, size_t ws_size,
                              hipStream_t stream)
{
    (void)n_in; (void)d_ws; (void)ws_size; (void)out_size;

    const int*   row    = (const int*)  d_in[0];
    const int*   col    = (const int*)  d_in[1];
    const float* dij    = (const float*)d_in[2];
    const float* charge = (const float*)d_in[3];
    const float* c6     = (const float*)d_in[4];
    const float* r0     = (const float*)d_in[5];

    const int E = in_sizes[0];   // number of edges

    float* out       = (float*)d_out;
    float* out_ecoul = out;                 // [E]
    float* out_cfij  = out + (size_t)E;     // [E,3]
    float* out_edisp = out + (size_t)4 * E; // [E]
    float* out_dfij  = out + (size_t)5 * E; // [E,3]

    const int block = 256;                  // 8 wave32 per block
    const int grid  = (E + block - 1) / block;

    bamboo_edge_kernel<<<grid, block, 0, stream>>>(
        row, col, dij, charge, c6, r0,
        out_ecoul, out_cfij, out_edisp, out_dfij, E);
}